// LIF_56573309224405
// MI455X (gfx1250) — compile-verified
//
#include <hip/hip_runtime.h>

// LIF forward scan over T=32 timesteps.
// x, out: [T, S] with S = B*C*H*W = 4,194,304 fp32 elements (16 MiB/timestep).
// One thread owns 4 consecutive spatial elements (128-bit vector) and carries
// v across T. Memory-bound: ~1.07 GB total traffic -> ~46 us floor @ 23.3 TB/s.

#define T_STEPS 32

// Native clang vector type: valid operand for __builtin_nontemporal_load/store
// (HIP's float4 is a struct and is rejected by the builtin).
typedef float v4f __attribute__((ext_vector_type(4)));

__global__ __launch_bounds__(256) void lif_fwd_kernel(
    const v4f* __restrict__ x, v4f* __restrict__ out, int s4) {
  const int i = blockIdx.x * blockDim.x + threadIdx.x;
  if (i >= s4) return;

  const v4f* xp = x + i;    // column i, row stride s4 (timestep stride)
  v4f*       op = out + i;

  v4f v = (v4f)(0.0f);

#pragma unroll 4
  for (int t = 0; t < T_STEPS; ++t) {
    // gfx1250 global_prefetch_b8: pull the t+4 timestep's line toward L2.
    // Wave-uniform test; under unroll 4, t is not a compile-time constant.
    if (t + 4 < T_STEPS) {
      __builtin_prefetch((const void*)(xp + (size_t)(t + 4) * s4), 0, 1);
    }

    // Touch-once streaming data: non-temporal (TH=NT) 128-bit load/store so
    // 1.07 GB of one-shot traffic doesn't rinse the 192 MB L2.
    const v4f xt = __builtin_nontemporal_load(xp + (size_t)t * s4);

    v4f sp;
#pragma unroll
    for (int k = 0; k < 4; ++k) {
      float vk = v[k];
      // v += (x - v)/TAU, TAU=2.0 -> (x-v)*0.5 exact, fma rounds once:
      // bit-identical to reference v + (x_t - v)/2.0
      vk = fmaf(xt[k] - vk, 0.5f, vk);
      const bool fire = (vk >= 1.0f);  // == heaviside(v - 1.0)
      sp[k] = fire ? 1.0f : 0.0f;
      v[k]  = fire ? 0.0f : vk;        // hard reset
    }

    __builtin_nontemporal_store(sp, op + (size_t)t * s4);
  }
}

extern "C" void kernel_launch(void* const* d_in, const int* in_sizes, int n_in,
                              void* d_out, int out_size, void* d_ws, size_t ws_size,
                              hipStream_t stream) {
  (void)n_in; (void)out_size; (void)d_ws; (void)ws_size;

  const float* x  = (const float*)d_in[0];
  float*      out = (float*)d_out;

  const long long total   = in_sizes[0];          // T * B*C*H*W = 134,217,728
  const long long spatial = total / T_STEPS;      // 4,194,304 elements/timestep
  const int s4 = (int)(spatial / 4);              // 1,048,576 v4f lanes

  const int threads = 256;                        // 8 wave32 waves per block
  const int blocks  = (s4 + threads - 1) / threads;

  lif_fwd_kernel<<<blocks, threads, 0, stream>>>(
      (const v4f*)x, (v4f*)out, s4);
}